// ASTSN_Weight_36601711296536
// MI455X (gfx1250) — compile-verified
//
#include <hip/hip_runtime.h>

typedef __attribute__((ext_vector_type(2))) float v2f;
typedef __attribute__((ext_vector_type(8))) float v8f;

#define BATCH 8
#define IN_CH 64
#define CWCH  177   // 2*64 + 49

#define WMMA_F32(a, b, c) \
  __builtin_amdgcn_wmma_f32_16x16x4_f32(false, (a), false, (b), (short)0, (c), false, false)

// ---------------------------------------------------------------------------
// 2x2 average pool: in (B,C,2H,2W) -> out (B,C,H,W)
// ---------------------------------------------------------------------------
__global__ void pool2_kernel(const float* __restrict__ in, float* __restrict__ out,
                             int total, int C, int Hout, int Wout) {
  int idx = blockIdx.x * blockDim.x + threadIdx.x;
  if (idx >= total) return;
  int w = idx % Wout; int t = idx / Wout;
  int h = t % Hout;  t /= Hout;
  int c = t % C;     int b = t / C;
  int Win = Wout * 2;
  const float* p = in + (((size_t)(b * C + c) * (Hout * 2) + h * 2) * Win + w * 2);
  out[idx] = 0.25f * (p[0] + p[1] + p[Win] + p[Win + 1]);
}

// ---------------------------------------------------------------------------
// Build feat (B,177,64,80): ch 0..48 = 49-shift correlation of pooled x0/y0,
// ch 49..112 = x, ch 113..176 = y
// ---------------------------------------------------------------------------
__global__ void corr_feat_kernel(const float* __restrict__ x0p, const float* __restrict__ y0p,
                                 const float* __restrict__ x,   const float* __restrict__ y,
                                 float* __restrict__ feat) {
  const int H = 64, W = 80;
  int idx = blockIdx.x * blockDim.x + threadIdx.x;
  int total = BATCH * CWCH * H * W;
  if (idx >= total) return;
  int w = idx % W; int t = idx / W;
  int h = t % H;   t /= H;
  int c = t % CWCH; int b = t / CWCH;
  float v;
  if (c < 49) {
    int dy = -6 + 2 * (c / 7);
    int dx = -6 + 2 * (c % 7);
    int yh = h + dy, yw = w + dx;
    float acc = 0.f;
    if (yh >= 0 && yh < H && yw >= 0 && yw < W) {
      const float* xp = x0p + (((size_t)b * IN_CH) * H + h)  * W + w;
      const float* yp = y0p + (((size_t)b * IN_CH) * H + yh) * W + yw;
      for (int cc = 0; cc < IN_CH; ++cc)
        acc += xp[(size_t)cc * H * W] * yp[(size_t)cc * H * W];
    }
    v = acc * (1.f / 64.f);
  } else if (c < 49 + IN_CH) {
    v = x[(((size_t)b * IN_CH + (c - 49)) * H + h) * W + w];
  } else {
    v = y[(((size_t)b * IN_CH + (c - 113)) * H + h) * W + w];
  }
  feat[idx] = v;
}

// ---------------------------------------------------------------------------
// Implicit-GEMM convolution via V_WMMA_F32_16X16X4_F32 with 2x2 register
// tiling: one wave32 computes a 32(row)x32(col) block of
//   C[co, pixel] = sum_k W[co,k] * im2col[k,pixel],   k = (ci, kh, kw)
// Reduction iterates (kh, kw) outer / ci inner so the inner loop is pure
// strided pointer arithmetic (no div/mod) and issues 4 wmma per 4-wide step.
// perBatchW=1 selects per-sample weights (adaptive conv).
// ---------------------------------------------------------------------------
__global__ void conv_wmma_kernel(const float* __restrict__ in, const float* __restrict__ wgt,
                                 float* __restrict__ out,
                                 int Cin, int Hin, int Win,
                                 int Cout, int Hout, int Wout,
                                 int Kdim, int stride, int pad,
                                 int doRelu, int perBatchW) {
  const int KK   = Kdim * Kdim;
  const int Ktot = Cin * KK;
  const int M    = Cout;
  const int N    = Hout * Wout;
  const int MT2  = (M + 31) >> 5;   // 32-row super-tiles
  const int NT2  = (N + 31) >> 5;   // 32-col super-tiles
  const int TPB  = MT2 * NT2;

  int wave = (blockIdx.x * blockDim.x + threadIdx.x) >> 5;
  if (wave >= BATCH * TPB) return;
  int b  = wave / TPB;
  int t  = wave % TPB;
  int tm = t / NT2, tn = t % NT2;

  int lane = threadIdx.x & 31;
  int half = lane >> 4;     // 0: lanes 0-15, 1: lanes 16-31
  int l15  = lane & 15;

  const float* Wb  = wgt + (perBatchW ? (size_t)b * M * Ktot : (size_t)0);
  const float* inb = in + (size_t)b * Cin * Hin * Win;
  const int HW = Hin * Win;

  // A rows (two 16-row tiles)
  int m0 = tm * 32 + l15;
  int m1 = m0 + 16;
  bool mv0 = m0 < M, mv1 = m1 < M;
  const float* ar0 = Wb + (size_t)(mv0 ? m0 : 0) * Ktot;
  const float* ar1 = Wb + (size_t)(mv1 ? m1 : 0) * Ktot;

  // B/C columns (two 16-col tiles)
  int n0 = tn * 32 + l15;
  int n1 = n0 + 16;
  bool nv0 = n0 < N, nv1 = n1 < N;
  int oh0 = nv0 ? n0 / Wout : 0, ow0 = nv0 ? n0 % Wout : 0;
  int oh1 = nv1 ? n1 / Wout : 0, ow1 = nv1 ? n1 % Wout : 0;

  v8f acc00 = {0.f,0.f,0.f,0.f,0.f,0.f,0.f,0.f};
  v8f acc01 = acc00, acc10 = acc00, acc11 = acc00;

  for (int kh = 0; kh < Kdim; ++kh) {
    int ih0 = oh0 * stride + kh - pad;
    int ih1 = oh1 * stride + kh - pad;
    bool hv0 = (unsigned)ih0 < (unsigned)Hin;
    bool hv1 = (unsigned)ih1 < (unsigned)Hin;
    for (int kw = 0; kw < Kdim; ++kw) {
      int iw0 = ow0 * stride + kw - pad;
      int iw1 = ow1 * stride + kw - pad;
      bool pv0 = nv0 && hv0 && ((unsigned)iw0 < (unsigned)Win);
      bool pv1 = nv1 && hv1 && ((unsigned)iw1 < (unsigned)Win);
      const float* bp0 = inb + (pv0 ? (size_t)ih0 * Win + iw0 : 0);
      const float* bp1 = inb + (pv1 ? (size_t)ih1 * Win + iw1 : 0);
      const int koff = kh * Kdim + kw;
      const float* a0p = ar0 + koff;   // stride KK floats per ci
      const float* a1p = ar1 + koff;
#pragma unroll 4
      for (int ci0 = 0; ci0 < Cin; ci0 += 4) {
        int ca = ci0 + half * 2;       // ISA 16x4 layout: half-wave K split
        bool c0v = ca < Cin;
        bool c1v = (ca + 1) < Cin;
        v2f a0, a1, b0, b1;
        a0.x = (mv0 && c0v) ? a0p[(size_t)ca * KK]       : 0.f;
        a0.y = (mv0 && c1v) ? a0p[(size_t)(ca + 1) * KK] : 0.f;
        a1.x = (mv1 && c0v) ? a1p[(size_t)ca * KK]       : 0.f;
        a1.y = (mv1 && c1v) ? a1p[(size_t)(ca + 1) * KK] : 0.f;
        b0.x = (pv0 && c0v) ? bp0[(size_t)ca * HW]       : 0.f;
        b0.y = (pv0 && c1v) ? bp0[(size_t)(ca + 1) * HW] : 0.f;
        b1.x = (pv1 && c0v) ? bp1[(size_t)ca * HW]       : 0.f;
        b1.y = (pv1 && c1v) ? bp1[(size_t)(ca + 1) * HW] : 0.f;
        acc00 = WMMA_F32(a0, b0, acc00);
        acc01 = WMMA_F32(a0, b1, acc01);
        acc10 = WMMA_F32(a1, b0, acc10);
        acc11 = WMMA_F32(a1, b1, acc11);
      }
    }
  }

  float* outb = out + (size_t)b * M * Hout * Wout;
#pragma unroll
  for (int r = 0; r < 8; ++r) {
    int r0 = tm * 32 + half * 8 + r;   // C/D layout: VGPR r -> M=r (lo) / r+8 (hi)
    int r1 = r0 + 16;
    if (r0 < M) {
      if (nv0) {
        float v = acc00[r]; if (doRelu) v = fmaxf(v, 0.f);
        outb[((size_t)r0 * Hout + oh0) * Wout + ow0] = v;
      }
      if (nv1) {
        float v = acc01[r]; if (doRelu) v = fmaxf(v, 0.f);
        outb[((size_t)r0 * Hout + oh1) * Wout + ow1] = v;
      }
    }
    if (r1 < M) {
      if (nv0) {
        float v = acc10[r]; if (doRelu) v = fmaxf(v, 0.f);
        outb[((size_t)r1 * Hout + oh0) * Wout + ow0] = v;
      }
      if (nv1) {
        float v = acc11[r]; if (doRelu) v = fmaxf(v, 0.f);
        outb[((size_t)r1 * Hout + oh1) * Wout + ow1] = v;
      }
    }
  }
}

// ---------------------------------------------------------------------------
// Spatial mean: in (B,C,HW) -> out (B,C)
// ---------------------------------------------------------------------------
__global__ void spatial_mean_kernel(const float* __restrict__ in, float* __restrict__ out,
                                    int C, int HW) {
  int idx = blockIdx.x * blockDim.x + threadIdx.x;
  if (idx >= BATCH * C) return;
  const float* p = in + (size_t)idx * HW;
  float acc = 0.f;
  for (int i = 0; i < HW; ++i) acc += p[i];
  out[idx] = acc / (float)HW;
}

// ---------------------------------------------------------------------------
// out[b,i,j] = fv[b,i] * ww[i,j] + wb[i,j]   (B,C,J)
// ---------------------------------------------------------------------------
__global__ void adaptive_weight_kernel(const float* __restrict__ fv, const float* __restrict__ ww,
                                       const float* __restrict__ wb, float* __restrict__ out,
                                       int C, int J) {
  int idx = blockIdx.x * blockDim.x + threadIdx.x;
  if (idx >= BATCH * C * J) return;
  int j = idx % J; int t = idx / J;
  int i = t % C;   int b = t / C;
  out[idx] = fv[b * C + i] * ww[(size_t)i * J + j] + wb[(size_t)i * J + j];
}

// ---------------------------------------------------------------------------
static inline int cdiv_h(int a, int b) { return (a + b - 1) / b; }

static inline void launch_conv(const float* in, const float* w, float* out,
                               int Cin, int Hin, int Win,
                               int Cout, int Hout, int Wout,
                               int Kdim, int stride, int pad,
                               int relu, int perBatch, hipStream_t stream) {
  int MT2 = cdiv_h(Cout, 32);
  int NT2 = cdiv_h(Hout * Wout, 32);
  int waves = BATCH * MT2 * NT2;
  int threads = waves * 32;
  int blocks = cdiv_h(threads, 256);
  conv_wmma_kernel<<<blocks, 256, 0, stream>>>(in, w, out, Cin, Hin, Win,
                                               Cout, Hout, Wout, Kdim, stride, pad,
                                               relu, perBatch);
}

extern "C" void kernel_launch(void* const* d_in, const int* in_sizes, int n_in,
                              void* d_out, int out_size, void* d_ws, size_t ws_size,
                              hipStream_t stream) {
  const float* x0    = (const float*)d_in[0];   // (8,64,128,160)
  const float* y0    = (const float*)d_in[1];
  const float* x     = (const float*)d_in[2];   // (8,64,64,80)
  const float* y     = (const float*)d_in[3];
  const float* w1a   = (const float*)d_in[4];   // (64,177,5,5)
  const float* w1b   = (const float*)d_in[5];   // (64,64,5,5)
  const float* w1c   = (const float*)d_in[6];   // (177,64,3,3)
  const float* w2a   = (const float*)d_in[7];
  const float* w2b   = (const float*)d_in[8];
  const float* w2c   = (const float*)d_in[9];   // (64,64,3,3)
  const float* wx_w  = (const float*)d_in[10];  // (177,1593)
  const float* wx_b  = (const float*)d_in[11];
  const float* wxf_w = (const float*)d_in[12];  // (64,576)
  const float* wxf_b = (const float*)d_in[13];
  float* out = (float*)d_out;                   // (8,64,64,3,3)

  // ---- workspace layout (floats) ----
  float* ws    = (float*)d_ws;
  float* x0p   = ws;                       // 8*64*64*80   = 2,621,440
  float* y0p   = x0p  + 2621440;           // 2,621,440
  float* feat  = y0p  + 2621440;           // 8*177*64*80  = 7,249,920
  float* h1    = feat + 7249920;           // 8*64*32*40   =   655,360
  float* h2    = h1   + 655360;            // 8*64*16*20   =   163,840
  float* h3    = h2   + 163840;            // 8*177*8*10   =   113,280
  float* fv    = h3   + 113280;            // 8*177        =     1,416
  float* wx    = fv   + 1416;              // 8*177*1593   = 2,255,688
  float* feat2 = wx   + 2255688;           // 7,249,920
  // branch 2 reuses h1/h2/h3/fv

  const int T256 = 256;

  // 1) pool x0, y0
  {
    int total = BATCH * IN_CH * 64 * 80;
    pool2_kernel<<<cdiv_h(total, T256), T256, 0, stream>>>(x0, x0p, total, IN_CH, 64, 80);
    pool2_kernel<<<cdiv_h(total, T256), T256, 0, stream>>>(y0, y0p, total, IN_CH, 64, 80);
  }

  // 2) corr + concat -> feat
  {
    int total = BATCH * CWCH * 64 * 80;
    corr_feat_kernel<<<cdiv_h(total, T256), T256, 0, stream>>>(x0p, y0p, x, y, feat);
  }

  // 3) branch 1
  launch_conv(feat, w1a, h1, CWCH, 64, 80,  64, 32, 40, 5, 2, 2, /*relu*/1, 0, stream);
  launch_conv(h1,   w1b, h2,   64, 32, 40,  64, 16, 20, 5, 2, 2, /*relu*/1, 0, stream);
  launch_conv(h2,   w1c, h3,   64, 16, 20, CWCH,  8, 10, 3, 2, 1, /*relu*/0, 0, stream);
  spatial_mean_kernel<<<cdiv_h(BATCH * CWCH, T256), T256, 0, stream>>>(h3, fv, CWCH, 80);

  // 4) per-sample adaptive filter wx = fv (x) wx_w + wx_b   (8,177,1593)
  {
    int total = BATCH * CWCH * 1593;
    adaptive_weight_kernel<<<cdiv_h(total, T256), T256, 0, stream>>>(fv, wx_w, wx_b, wx, CWCH, 1593);
  }

  // 5) adaptive conv (per-batch weights) + relu -> feat2
  launch_conv(feat, wx, feat2, CWCH, 64, 80, CWCH, 64, 80, 3, 1, 1, /*relu*/1, /*perBatch*/1, stream);

  // 6) branch 2
  launch_conv(feat2, w2a, h1, CWCH, 64, 80, 64, 32, 40, 5, 2, 2, /*relu*/1, 0, stream);
  launch_conv(h1,    w2b, h2,   64, 32, 40, 64, 16, 20, 5, 2, 2, /*relu*/1, 0, stream);
  launch_conv(h2,    w2c, h3,   64, 16, 20, 64,  8, 10, 3, 2, 1, /*relu*/0, 0, stream);
  spatial_mean_kernel<<<cdiv_h(BATCH * 64, T256), T256, 0, stream>>>(h3, fv, 64, 80);

  // 7) final output wxf = fv (x) wxf_w + wxf_b  -> (8,64,64,3,3)
  {
    int total = BATCH * 64 * 576;
    adaptive_weight_kernel<<<cdiv_h(total, T256), T256, 0, stream>>>(fv, wxf_w, wxf_b, out, 64, 576);
  }
}